// MPC_37675453120813
// MI455X (gfx1250) — compile-verified
//
#include <hip/hip_runtime.h>

#define NXD 36
#define NUD 12
#define NSCD 48
#define LDD 52      // LDS row stride for square work matrices
#define LDF 48      // packed stride for async-loaded F tiles (16B aligned rows)
#define NLS 10
#define REGU 1e-6f

typedef __attribute__((ext_vector_type(2))) float v2f;
typedef __attribute__((ext_vector_type(8))) float v8f;

__device__ __forceinline__ v8f wmma4(v2f a, v2f b, v8f c) {
  // V_WMMA_F32_16X16X4_F32 : D = A(16x4) * B(4x16) + C, full fp32
  return __builtin_amdgcn_wmma_f32_16x16x4_f32(false, a, false, b, (short)0, c,
                                               false, false);
}

// Async global->LDS copy of one F tile (36x48 fp32 = 432 x b128), ASYNCcnt.
__device__ __forceinline__ void async_copy_F(float* dst, const float* src,
                                             int lane) {
  const uint32_t base = (uint32_t)(uintptr_t)dst; // low 32 bits = LDS offset
  for (int i = lane; i < (NXD * NSCD) / 4; i += 32) {
    asm volatile("global_load_async_to_lds_b128 %0, %1, off"
                 :
                 : "v"(base + (uint32_t)i * 16u), "v"(src + i * 4)
                 : "memory");
  }
}

__device__ __forceinline__ void wait_async0() {
  asm volatile("s_wait_asynccnt 0x0" ::: "memory");
}

// D(48x48) = op(A) * op(B), contraction length = 4*KSTEPS, operands in LDS.
// Zero-padded regions make the 48-padding mathematically exact.
template <int KSTEPS, bool TA, bool TB>
__device__ __forceinline__ void mm48(const float* A, int lda, const float* Bm,
                                     int ldb, float* D, int ldd, int lane) {
  const int m = lane & 15;
  const int kb = (lane >> 4) * 2;   // A/B K-pair base per half-wave
  const int rofs = (lane >> 4) * 8; // C/D row offset per half-wave
  for (int ti = 0; ti < 3; ++ti) {
    const int r = ti * 16 + m;
    for (int tj = 0; tj < 3; ++tj) {
      const int n = tj * 16 + m;
      v8f acc = {0.f, 0.f, 0.f, 0.f, 0.f, 0.f, 0.f, 0.f};
#pragma unroll
      for (int kt = 0; kt < KSTEPS; ++kt) {
        const int k0 = kt * 4 + kb;
        v2f a, b;
        a.x = TA ? A[k0 * lda + r] : A[r * lda + k0];
        a.y = TA ? A[(k0 + 1) * lda + r] : A[r * lda + k0 + 1];
        b.x = TB ? Bm[n * ldb + k0] : Bm[k0 * ldb + n];
        b.y = TB ? Bm[n * ldb + k0 + 1] : Bm[(k0 + 1) * ldb + n];
        acc = wmma4(a, b, acc);
      }
#pragma unroll
      for (int i = 0; i < 8; ++i)
        D[(ti * 16 + rofs + i) * ldd + tj * 16 + m] = acc[i];
    }
  }
}

struct Smem {
  alignas(16) float Fm[2][NXD * LDF]; // double-buffered F_t (async target)
  float Vm[48 * LDD];   // value matrix V (padded 48x48, rows/cols>=36 zero)
  float FVm[48 * LDD];  // F^T V, later reused as Qxu*K temp
  float Qm[48 * LDD];   // Q (48x48)
  float Kp[12 * LDD];   // gain K (12x36, cols>=36 zero)
  float aug[12 * LDD];  // [Quu | Qux | qu] augmented solve workspace (12x49)
  float vvec[48], qvec[48], fvec[48], cvec[48], cdvec[48];
  float kvl[12], fac[12], uv[12];
  float X[NLS][48];     // rollout states for each alpha
  float costs[NLS];
  float xv[48];         // best-alpha rollout state
};

__global__ __launch_bounds__(32) void lqr_kernel(
    const float* __restrict__ x0, const float* __restrict__ Cd,
    const float* __restrict__ c, const float* __restrict__ F,
    const float* __restrict__ f, float* __restrict__ out,
    float* __restrict__ Kws, float* __restrict__ kvws, int T) {
  __shared__ Smem s;
  const int lane = threadIdx.x;
  const int b = blockIdx.x;

  { // zero all LDS once: establishes zero padding everywhere
    float* p = (float*)&s;
    const int n = (int)(sizeof(Smem) / 4);
    for (int i = lane; i < n; i += 32) p[i] = 0.f;
  }
  __syncthreads();

  const size_t bT = (size_t)b * (size_t)T;

  // ---------------- backward Riccati pass (async double-buffered F) -------
  int cur = 0;
  async_copy_F(s.Fm[0], F + (bT + (size_t)(T - 1)) * (NXD * NSCD), lane);
  for (int t = T - 1; t >= 0; --t) {
    const size_t bt = bT + (size_t)t;
    wait_async0(); // F[t] resident in Fm[cur]
    if (t > 0)     // overlap next (earlier) step's F with this step's compute
      async_copy_F(s.Fm[1 - cur], F + (bt - 1) * (size_t)(NXD * NSCD), lane);
    const float* Fl = s.Fm[cur];
    for (int i = lane; i < NSCD; i += 32) {
      s.cdvec[i] = Cd[bt * NSCD + i];
      s.cvec[i] = c[bt * NSCD + i];
    }
    for (int i = lane; i < NXD; i += 32) s.fvec[i] = f[bt * NXD + i];
    __syncthreads();

    // FV = F^T V   (K = 36 -> 9 wmma k-steps per tile)
    mm48<9, true, false>(Fl, LDF, s.Vm, LDD, s.FVm, LDD, lane);
    __syncthreads();

    // q = c + FV f + F^T v
    for (int si = lane; si < NSCD; si += 32) {
      float acc = s.cvec[si];
      for (int x = 0; x < NXD; ++x)
        acc += s.FVm[si * LDD + x] * s.fvec[x] + Fl[x * LDF + si] * s.vvec[x];
      s.qvec[si] = acc;
    }
    __syncthreads();

    // Q = FV F + diag(Cd) (+ REG on uu diagonal)
    mm48<9, false, false>(s.FVm, LDD, Fl, LDF, s.Qm, LDD, lane);
    __syncthreads();
    for (int d = lane; d < NSCD; d += 32)
      s.Qm[d * LDD + d] += s.cdvec[d] + (d >= NXD ? REGU : 0.f);
    __syncthreads();

    // augmented system [Quu | Qux | qu] (12x49)
    for (int i = lane; i < NUD * 49; i += 32) {
      const int r = i / 49, cc2 = i % 49;
      float v;
      if (cc2 < NUD) v = s.Qm[(NXD + r) * LDD + NXD + cc2];
      else if (cc2 < 48) v = s.Qm[(NXD + r) * LDD + (cc2 - NUD)];
      else v = s.qvec[NXD + r];
      s.aug[r * LDD + cc2] = v;
    }
    __syncthreads();

    // Gauss-Jordan (SPD + REG: no pivoting needed)
    for (int p = 0; p < NUD; ++p) {
      const float pinv = 1.0f / s.aug[p * LDD + p];
      for (int cc2 = lane; cc2 < 49; cc2 += 32) s.aug[p * LDD + cc2] *= pinv;
      __syncthreads();
      if (lane < NUD) s.fac[lane] = s.aug[lane * LDD + p];
      __syncthreads();
      for (int i = lane; i < NUD * 49; i += 32) {
        const int r = i / 49, cc2 = i % 49;
        if (r != p) s.aug[r * LDD + cc2] -= s.fac[r] * s.aug[p * LDD + cc2];
      }
      __syncthreads();
    }

    // K = -Quu^{-1} Qux, kv = -Quu^{-1} qu  -> LDS + workspace
    for (int i = lane; i < NUD * NXD; i += 32) {
      const int j = i / NXD, x = i % NXD;
      const float kvv = -s.aug[j * LDD + NUD + x];
      s.Kp[j * LDD + x] = kvv;
      Kws[bt * (size_t)(NUD * NXD) + i] = kvv;
    }
    if (lane < NUD) {
      const float kk = -s.aug[lane * LDD + 48];
      s.kvl[lane] = kk;
      kvws[bt * NUD + lane] = kk;
    }
    __syncthreads();

    // Tn = Qxu K (K = 12 -> 3 k-steps), reuse FVm
    mm48<3, false, false>(s.Qm + NXD, LDD, s.Kp, LDD, s.FVm, LDD, lane);
    __syncthreads();

    // V <- sym(Qxx + Qxu K), v <- qx + Qxu kv (padding stays zero)
    for (int i = lane; i < NSCD * NSCD; i += 32) {
      const int r = i / NSCD, cc2 = i % NSCD;
      float v = 0.f;
      if (r < NXD && cc2 < NXD)
        v = 0.5f * (s.Qm[r * LDD + cc2] + s.FVm[r * LDD + cc2] +
                    s.Qm[cc2 * LDD + r] + s.FVm[cc2 * LDD + r]);
      s.Vm[r * LDD + cc2] = v;
    }
    for (int x = lane; x < NSCD; x += 32) {
      float v = 0.f;
      if (x < NXD) {
        v = s.qvec[x];
        for (int j = 0; j < NUD; ++j) v += s.Qm[x * LDD + NXD + j] * s.kvl[j];
      }
      s.vvec[x] = v;
    }
    __syncthreads();
    cur ^= 1;
  }

  // ---------------- rollout phase 1: cost for all 10 alphas ----------------
  float al[NLS];
  al[0] = 1.f;
  for (int a = 1; a < NLS; ++a) al[a] = al[a - 1] * 0.2f;
  float costp[NLS];
  for (int a = 0; a < NLS; ++a) costp[a] = 0.f;
  for (int a = 0; a < NLS; ++a)
    for (int si = lane; si < NXD; si += 32)
      s.X[a][si] = x0[(size_t)b * NXD + si];
  __syncthreads();

  cur = 0;
  async_copy_F(s.Fm[0], F + bT * (size_t)(NXD * NSCD), lane);
  for (int t = 0; t < T; ++t) {
    const size_t bt = bT + (size_t)t;
    wait_async0();
    if (t + 1 < T)
      async_copy_F(s.Fm[1 - cur], F + (bt + 1) * (size_t)(NXD * NSCD), lane);
    const float* Fl = s.Fm[cur];
    for (int i = lane; i < NSCD; i += 32) {
      s.cdvec[i] = Cd[bt * NSCD + i];
      s.cvec[i] = c[bt * NSCD + i];
    }
    for (int i = lane; i < NXD; i += 32) s.fvec[i] = f[bt * NXD + i];
    for (int i = lane; i < NUD * NXD; i += 32)
      s.Kp[(i / NXD) * LDD + (i % NXD)] = Kws[bt * (size_t)(NUD * NXD) + i];
    if (lane < NUD) s.kvl[lane] = kvws[bt * NUD + lane];
    __syncthreads();

    for (int a = 0; a < NLS; ++a) {
      if (lane < NUD) {
        float u = al[a] * s.kvl[lane];
        for (int x = 0; x < NXD; ++x) u += s.Kp[lane * LDD + x] * s.X[a][x];
        s.uv[lane] = u;
      }
      __syncthreads();
      for (int si = lane; si < NSCD; si += 32) {
        const float tau = (si < NXD) ? s.X[a][si] : s.uv[si - NXD];
        costp[a] += (0.5f * s.cdvec[si] * tau + s.cvec[si]) * tau;
      }
      float xn0 = 0.f, xn1 = 0.f;
      if (lane < NXD) {
        xn0 = s.fvec[lane];
        for (int si = 0; si < NSCD; ++si) {
          const float tau = (si < NXD) ? s.X[a][si] : s.uv[si - NXD];
          xn0 += Fl[lane * LDF + si] * tau;
        }
      }
      if (lane + 32 < NXD) {
        xn1 = s.fvec[lane + 32];
        for (int si = 0; si < NSCD; ++si) {
          const float tau = (si < NXD) ? s.X[a][si] : s.uv[si - NXD];
          xn1 += Fl[(lane + 32) * LDF + si] * tau;
        }
      }
      __syncthreads();
      if (lane < NXD) s.X[a][lane] = xn0;
      if (lane + 32 < NXD) s.X[a][lane + 32] = xn1;
      __syncthreads();
    }
    cur ^= 1;
  }

  for (int a = 0; a < NLS; ++a) { // wave32 cost reduction
    float v = costp[a];
    for (int off = 16; off > 0; off >>= 1) v += __shfl_xor(v, off, 32);
    if (lane == 0) s.costs[a] = v;
  }
  __syncthreads();
  int bi = 0;
  float bc = s.costs[0];
  for (int a = 1; a < NLS; ++a)
    if (s.costs[a] < bc) { bc = s.costs[a]; bi = a; }
  const float ab = al[bi];

  // ---------------- rollout phase 2: replay best alpha, emit tau ----------
  for (int si = lane; si < NXD; si += 32) s.xv[si] = x0[(size_t)b * NXD + si];
  __syncthreads();
  cur = 0;
  async_copy_F(s.Fm[0], F + bT * (size_t)(NXD * NSCD), lane);
  for (int t = 0; t < T; ++t) {
    const size_t bt = bT + (size_t)t;
    wait_async0();
    if (t + 1 < T)
      async_copy_F(s.Fm[1 - cur], F + (bt + 1) * (size_t)(NXD * NSCD), lane);
    const float* Fl = s.Fm[cur];
    for (int i = lane; i < NXD; i += 32) s.fvec[i] = f[bt * NXD + i];
    for (int i = lane; i < NUD * NXD; i += 32)
      s.Kp[(i / NXD) * LDD + (i % NXD)] = Kws[bt * (size_t)(NUD * NXD) + i];
    if (lane < NUD) s.kvl[lane] = kvws[bt * NUD + lane];
    __syncthreads();

    if (lane < NUD) {
      float u = ab * s.kvl[lane];
      for (int x = 0; x < NXD; ++x) u += s.Kp[lane * LDD + x] * s.xv[x];
      s.uv[lane] = u;
    }
    __syncthreads();
    for (int si = lane; si < NSCD; si += 32)
      out[bt * NSCD + si] = (si < NXD) ? s.xv[si] : s.uv[si - NXD];
    float xn0 = 0.f, xn1 = 0.f;
    if (lane < NXD) {
      xn0 = s.fvec[lane];
      for (int si = 0; si < NSCD; ++si) {
        const float tau = (si < NXD) ? s.xv[si] : s.uv[si - NXD];
        xn0 += Fl[lane * LDF + si] * tau;
      }
    }
    if (lane + 32 < NXD) {
      xn1 = s.fvec[lane + 32];
      for (int si = 0; si < NSCD; ++si) {
        const float tau = (si < NXD) ? s.xv[si] : s.uv[si - NXD];
        xn1 += Fl[(lane + 32) * LDF + si] * tau;
      }
    }
    __syncthreads();
    if (lane < NXD) s.xv[lane] = xn0;
    if (lane + 32 < NXD) s.xv[lane + 32] = xn1;
    __syncthreads();
  }
}

extern "C" void kernel_launch(void* const* d_in, const int* in_sizes, int n_in,
                              void* d_out, int out_size, void* d_ws,
                              size_t ws_size, hipStream_t stream) {
  const float* x0 = (const float*)d_in[0];
  const float* Cd = (const float*)d_in[1];
  const float* c = (const float*)d_in[2];
  const float* F = (const float*)d_in[3];
  const float* f = (const float*)d_in[4];
  float* out = (float*)d_out;
  const int B = in_sizes[0] / NXD;                       // x0: (B, 36)
  const int T = (int)((size_t)in_sizes[4] / ((size_t)B * NXD)); // f: (B,T,36)
  float* Kws = (float*)d_ws;                             // B*T*12*36 floats
  float* kvws = Kws + (size_t)B * (size_t)T * NUD * NXD; // B*T*12 floats
  (void)n_in; (void)out_size; (void)ws_size;
  lqr_kernel<<<B, 32, 0, stream>>>(x0, Cd, c, F, f, out, Kws, kvws, T);
}